// NetGATDeform_64501818851755
// MI455X (gfx1250) — compile-verified
//
#include <hip/hip_runtime.h>
#include <hip/hip_bf16.h>
#include <stdint.h>

#define NN 10000
#define NE 70000
#define HH 6
#define NEG_SLOPE 0.2f

typedef __attribute__((ext_vector_type(2))) float v2f;
typedef __attribute__((ext_vector_type(8))) float v8f;

#define WMMA_F32(a, b, c) \
  __builtin_amdgcn_wmma_f32_16x16x4_f32(false, (a), false, (b), (short)0, (c), false, false)

// ---------------------------------------------------------------- device utils
__device__ __forceinline__ float selu_f(float x) {
  const float scale = 1.0507009873554805f;
  const float alph  = 1.6732632423543772f;
  return x > 0.f ? scale * x : scale * alph * (expf(x) - 1.f);
}

__device__ __forceinline__ void atomicMaxFloat(float* addr, float val) {
  int* ia = (int*)addr;
  int old = __float_as_int(*addr);
  while (val > __int_as_float(old)) {
    int assumed = old;
    old = atomicCAS(ia, assumed, __float_as_int(val));
    if (old == assumed) break;
  }
}

// ---------------------------------------------------------------- WMMA GEMM
// C[M x Np] = A[M x K] @ B[K x Np], fp32 via V_WMMA_F32_16X16X4_F32.
// B is pre-packed k-pair interleaved: (k,j) -> Bp[((k>>1)*Np + j)*2 + (k&1)],
// so each lane's B fragment {B[k][col], B[k+1][col]} is ONE b64 load, and the
// 4 column tiles are immediate offsets. One wave32 = 16x64 output strip
// (4 accumulators sharing one A fragment); block = 4 waves = 16x256.
// Register double-buffering: next k-step's 5 b64 loads are issued before the
// current step's 4 WMMAs so waits are partial, not full drains.
// A frag (16x4): VGPR0 = K0 (lanes 0-15)/K2 (lanes 16-31), VGPR1 = K1/K3.
// D (16x16):     VGPR r -> M=r (lanes 0-15), M=r+8 (lanes 16-31).
__global__ void __launch_bounds__(128)
gemm_f32_wmma_k(const float* __restrict__ A, const float* __restrict__ Bp,
                float* __restrict__ C, int K, int Np) {
  const int lane = threadIdx.x & 31;
  const int wave = threadIdx.x >> 5;             // 0..3
  const int hv   = lane >> 4;                    // lane half
  const int l16  = lane & 15;
  const int colBase = blockIdx.x * 256 + wave * 64;
  if (colBase >= Np) return;                     // wave-uniform exit
  const int row = blockIdx.y * 16 + l16;         // M = 10000 (multiple of 16)
  const int col = colBase + l16;
  const float* __restrict__ arow = A + (size_t)row * K;
  const float* __restrict__ brow = Bp + ((size_t)hv * Np + col) * 2;

  v8f acc0 = {}, acc1 = {}, acc2 = {}, acc3 = {};
  v2f a  = *(const v2f*)(arow + hv * 2);
  v2f b0 = *(const v2f*)(brow);
  v2f b1 = *(const v2f*)(brow + 32);
  v2f b2 = *(const v2f*)(brow + 64);
  v2f b3 = *(const v2f*)(brow + 96);

  for (int k0 = 4; k0 < K; k0 += 4) {
    // prefetch next k-step (5 independent b64 loads)
    const v2f an = *(const v2f*)(arow + k0 + hv * 2);
    const float* __restrict__ bn = brow + (size_t)(k0 >> 1) * Np * 2;
    const v2f bn0 = *(const v2f*)(bn);
    const v2f bn1 = *(const v2f*)(bn + 32);
    const v2f bn2 = *(const v2f*)(bn + 64);
    const v2f bn3 = *(const v2f*)(bn + 96);
    // compute on current fragments while prefetch is in flight
    acc0 = WMMA_F32(a, b0, acc0);
    acc1 = WMMA_F32(a, b1, acc1);
    acc2 = WMMA_F32(a, b2, acc2);
    acc3 = WMMA_F32(a, b3, acc3);
    a = an; b0 = bn0; b1 = bn1; b2 = bn2; b3 = bn3;
  }
  acc0 = WMMA_F32(a, b0, acc0);
  acc1 = WMMA_F32(a, b1, acc1);
  acc2 = WMMA_F32(a, b2, acc2);
  acc3 = WMMA_F32(a, b3, acc3);

  const int r0 = blockIdx.y * 16 + hv * 8;
  float* __restrict__ crow = C + (size_t)r0 * Np + col;
  #pragma unroll
  for (int r = 0; r < 8; ++r) {
    float* cr = crow + (size_t)r * Np;
    cr[0]  = acc0[r];
    cr[16] = acc1[r];
    cr[32] = acc2[r];
    cr[48] = acc3[r];
  }
}

// pad + pack W [K x Hd] -> Wp: (k,j) -> Wp[((k>>1)*Np + j)*2 + (k&1)]
__global__ void pad_w_k(const float* __restrict__ W, float* __restrict__ Wp,
                        int K, int Hd, int Np) {
  int i = blockIdx.x * blockDim.x + threadIdx.x;
  if (i >= K * Np) return;
  int k = i / Np, j = i % Np;
  float v = (j < Hd) ? W[(size_t)k * Hd + j] : 0.f;
  Wp[((size_t)(k >> 1) * Np + j) * 2 + (k & 1)] = v;
}

// ---------------------------------------------------------------- misc kernels
__global__ void fill_k(float* __restrict__ p, float v, int n) {
  int i = blockIdx.x * blockDim.x + threadIdx.x;
  if (i < n) p[i] = v;
}

__global__ void copy_cols_k(float* __restrict__ dst, int dld, int doff,
                            const float* __restrict__ src, int sld, int soff,
                            int n, int w) {
  int i = blockIdx.x * blockDim.x + threadIdx.x;
  if (i >= n * w) return;
  int node = i / w, j = i % w;
  dst[(size_t)node * dld + doff + j] = src[(size_t)node * sld + soff + j];
}

// h = selu(data @ W_lin + b); written into t1 columns [t_off, t_off+dlin)
__global__ void lin_selu_k(const float* __restrict__ data,
                           const float* __restrict__ W, const float* __restrict__ b,
                           float* __restrict__ t, int n, int din, int dlin,
                           int t_ld, int t_off) {
  int i = blockIdx.x * blockDim.x + threadIdx.x;
  if (i >= n * dlin) return;
  int node = i / dlin, j = i % dlin;
  float s = b[j];
  for (int k = 0; k < din; ++k) s += data[node * din + k] * W[k * dlin + j];
  t[(size_t)node * t_ld + t_off + j] = selu_f(s);
}

// per-node, per-head attention logits: one wave per head, block = 6 waves
__global__ void __launch_bounds__(HH * 32)
attn_dots_k(const float* __restrict__ x, int ldx,
            const float* __restrict__ asrc, const float* __restrict__ adst,
            float* __restrict__ out_s, float* __restrict__ out_d, int dout) {
  const int node = blockIdx.x;
  const int h = threadIdx.x >> 5;
  const int lane = threadIdx.x & 31;
  const float* xp = x + (size_t)node * ldx + (size_t)h * dout;
  const float* as = asrc + h * dout;
  const float* ad = adst + h * dout;
  float ss = 0.f, sd = 0.f;
  for (int c = lane; c < dout; c += 32) {
    float v = xp[c];
    ss += v * as[c];
    sd += v * ad[c];
  }
  for (int off = 16; off; off >>= 1) {
    ss += __shfl_down(ss, off, 32);
    sd += __shfl_down(sd, off, 32);
  }
  if (lane == 0) { out_s[node * HH + h] = ss; out_d[node * HH + h] = sd; }
}

__global__ void edge_e_k(const float* __restrict__ as, const float* __restrict__ ad,
                         const int* __restrict__ src, const int* __restrict__ dst,
                         float* __restrict__ ebuf, float* __restrict__ m) {
  int i = blockIdx.x * blockDim.x + threadIdx.x;
  if (i >= NE * HH) return;
  int e = i % NE, h = i / NE;
  int s = src[e], d = dst[e];
  float v = as[s * HH + h] + ad[d * HH + h];
  v = v > 0.f ? v : NEG_SLOPE * v;
  ebuf[(size_t)h * NE + e] = v;
  atomicMaxFloat(&m[d * HH + h], v);
}

__global__ void fix_m_k(float* __restrict__ m, int n) {
  int i = blockIdx.x * blockDim.x + threadIdx.x;
  if (i < n && m[i] < -1e29f) m[i] = 0.f;
}

__global__ void edge_ex_k(float* __restrict__ ebuf, const float* __restrict__ m,
                          const int* __restrict__ dst, float* __restrict__ den) {
  int i = blockIdx.x * blockDim.x + threadIdx.x;
  if (i >= NE * HH) return;
  int e = i % NE, h = i / NE;
  int d = dst[e];
  float ex = expf(ebuf[(size_t)h * NE + e] - m[d * HH + h]);
  ebuf[(size_t)h * NE + e] = ex;
  atomicAdd(&den[d * HH + h], ex);
}

__global__ void edge_alpha_k(float* __restrict__ ebuf, const float* __restrict__ den,
                             const int* __restrict__ dst) {
  int i = blockIdx.x * blockDim.x + threadIdx.x;
  if (i >= NE * HH) return;
  int e = i % NE, h = i / NE;
  float dv = den[dst[e] * HH + h];
  ebuf[(size_t)h * NE + e] /= (dv > 0.f ? dv : 1.f);
}

__global__ void edge_coord_k(const float* __restrict__ alpha,
                             const float* __restrict__ coord,
                             const int* __restrict__ src, const int* __restrict__ dst,
                             float* __restrict__ ncoord) {
  int e = blockIdx.x * blockDim.x + threadIdx.x;
  if (e >= NE) return;
  float am = 0.f;
  #pragma unroll
  for (int h = 0; h < HH; ++h) am += alpha[(size_t)h * NE + e];
  am *= (1.f / HH);
  int s = src[e], d = dst[e];
  atomicAdd(&ncoord[d * 2 + 0], am * coord[s * 2 + 0]);
  atomicAdd(&ncoord[d * 2 + 1], am * coord[s * 2 + 1]);
}

__global__ void edge_agg_k(const float* __restrict__ x, int ldx,
                           const float* __restrict__ alpha,
                           const int* __restrict__ src, const int* __restrict__ dst,
                           float* __restrict__ agg, int dout) {
  const size_t total = (size_t)NE * HH * dout;
  const size_t stride = (size_t)gridDim.x * blockDim.x;
  for (size_t i = (size_t)blockIdx.x * blockDim.x + threadIdx.x; i < total; i += stride) {
    int c = (int)(i % dout);
    size_t t = i / dout;
    int h = (int)(t % HH);
    int e = (int)(t / HH);
    float a = alpha[(size_t)h * NE + e];
    int s = src[e], d = dst[e];
    atomicAdd(&agg[(size_t)d * HH * dout + (size_t)h * dout + c],
              a * x[(size_t)s * ldx + (size_t)h * dout + c]);
  }
}

__global__ void node_out_k(const float* __restrict__ agg, float* __restrict__ f,
                           int n, int dout) {
  int i = blockIdx.x * blockDim.x + threadIdx.x;
  if (i >= n * dout) return;
  int node = i / dout, c = i % dout;
  const float* p = agg + (size_t)node * HH * dout + c;
  float s = 0.f;
  #pragma unroll
  for (int h = 0; h < HH; ++h) s += p[(size_t)h * dout];
  f[(size_t)node * dout + c] = selu_f(s * (1.f / HH));
}

__global__ void coord_out_k(const float* __restrict__ coord,
                            const float* __restrict__ ncoord,
                            const unsigned char* __restrict__ bd,
                            float* __restrict__ out, int n) {
  int i = blockIdx.x * blockDim.x + threadIdx.x;
  if (i >= n) return;
  bool b = bd[i] != 0;
  out[i * 2 + 0] = b ? coord[i * 2 + 0] : ncoord[i * 2 + 0];
  out[i * 2 + 1] = b ? coord[i * 2 + 1] : ncoord[i * 2 + 1];
}

// ---------------------------------------------------------------- host side
static inline int cdiv(int a, int b) { return (a + b - 1) / b; }

static void run_gat_layer(const float* tin, int din, const float* W,
                          const float* asrc, const float* adst, int dout,
                          const float* coord_in, float* coord_out, float* feat_out,
                          bool need_feat,
                          float* wpbuf, float* xbuf, float* aggbuf,
                          float* asb, float* adb, float* mbuf, float* denbuf,
                          float* ebuf, float* ncoord,
                          const int* src, const int* dst, const unsigned char* bd,
                          hipStream_t stream) {
  const int n = NN, H = HH, Hd = H * dout;
  const int Np = cdiv(Hd, 64) * 64;       // padded GEMM N, multiple of 64
  // 0. pad + pack weights (k-pair interleaved, zero-filled edge columns)
  pad_w_k<<<cdiv(din * Np, 256), 256, 0, stream>>>(W, wpbuf, din, Hd, Np);
  // 1. x = tin @ Wp  (WMMA fp32, branch-free, double-buffered)
  dim3 gg(cdiv(Np, 256), n / 16);
  gemm_f32_wmma_k<<<gg, 128, 0, stream>>>(tin, wpbuf, xbuf, din, Np);
  // 2. attention logits per node/head
  attn_dots_k<<<n, H * 32, 0, stream>>>(xbuf, Np, asrc, adst, asb, adb, dout);
  // 3. init segment buffers
  fill_k<<<cdiv(n * H, 256), 256, 0, stream>>>(mbuf, -1e30f, n * H);
  fill_k<<<cdiv(n * H, 256), 256, 0, stream>>>(denbuf, 0.f, n * H);
  fill_k<<<cdiv(n * 2, 256), 256, 0, stream>>>(ncoord, 0.f, n * 2);
  if (need_feat)
    fill_k<<<cdiv(n * Hd, 256), 256, 0, stream>>>(aggbuf, 0.f, n * Hd);
  // 4. edge softmax
  const int eh = NE * H;
  edge_e_k<<<cdiv(eh, 256), 256, 0, stream>>>(asb, adb, src, dst, ebuf, mbuf);
  fix_m_k<<<cdiv(n * H, 256), 256, 0, stream>>>(mbuf, n * H);
  edge_ex_k<<<cdiv(eh, 256), 256, 0, stream>>>(ebuf, mbuf, dst, denbuf);
  edge_alpha_k<<<cdiv(eh, 256), 256, 0, stream>>>(ebuf, denbuf, dst);
  // 5. coordinate update + feature aggregation
  edge_coord_k<<<cdiv(NE, 256), 256, 0, stream>>>(ebuf, coord_in, src, dst, ncoord);
  if (need_feat) {
    edge_agg_k<<<8192, 256, 0, stream>>>(xbuf, Np, ebuf, src, dst, aggbuf, dout);
    node_out_k<<<cdiv(n * dout, 256), 256, 0, stream>>>(aggbuf, feat_out, n, dout);
  }
  coord_out_k<<<cdiv(n, 256), 256, 0, stream>>>(coord_in, ncoord, bd, coord_out, n);
}

extern "C" void kernel_launch(void* const* d_in, const int* in_sizes, int n_in,
                              void* d_out, int out_size, void* d_ws, size_t ws_size,
                              hipStream_t stream) {
  const float* data  = (const float*)d_in[0];
  const int*   eidx  = (const int*)d_in[1];
  const unsigned char* bd = (const unsigned char*)d_in[2];  // jax bool -> 1 byte
  const float* W_lin = (const float*)d_in[4];
  const float* b_lin = (const float*)d_in[5];
  const float* W1 = (const float*)d_in[6];
  const float* a1s = (const float*)d_in[7];
  const float* a1d = (const float*)d_in[8];
  const float* W2 = (const float*)d_in[9];
  const float* a2s = (const float*)d_in[10];
  const float* a2d = (const float*)d_in[11];
  const float* W3 = (const float*)d_in[12];
  const float* a3s = (const float*)d_in[13];
  const float* a3d = (const float*)d_in[14];
  const float* W4 = (const float*)d_in[15];
  const float* a4s = (const float*)d_in[16];
  const float* a4d = (const float*)d_in[17];

  const int* src = eidx;
  const int* dst = eidx + NE;

  // workspace layout (floats)
  float* ws = (float*)d_ws;
  size_t off = 0;
  float* coords0 = ws + off; off += (size_t)NN * 2;
  float* c1      = ws + off; off += (size_t)NN * 2;
  float* c2      = ws + off; off += (size_t)NN * 2;
  float* c3      = ws + off; off += (size_t)NN * 2;
  float* tbuf    = ws + off; off += (size_t)NN * 512;
  float* xbuf    = ws + off; off += (size_t)NN * 3072;   // padded ld (max Np)
  float* aggbuf  = ws + off; off += (size_t)NN * 3048;
  float* fbuf    = ws + off; off += (size_t)NN * 508;
  float* wpbuf   = ws + off; off += (size_t)512 * 1536;  // max K*Np over layers
  float* asb     = ws + off; off += (size_t)NN * HH;
  float* adb     = ws + off; off += (size_t)NN * HH;
  float* mbuf    = ws + off; off += (size_t)NN * HH;
  float* denbuf  = ws + off; off += (size_t)NN * HH;
  float* ebuf    = ws + off; off += (size_t)NE * HH;
  float* ncoord  = ws + off; off += (size_t)NN * 2;

  // coords0 = data[:, 0:2]
  copy_cols_k<<<cdiv(NN * 2, 256), 256, 0, stream>>>(coords0, 2, 0, data, 10, 0, NN, 2);
  // t1 = [coords, selu(data @ W_lin + b_lin)]  (width 256)
  copy_cols_k<<<cdiv(NN * 2, 256), 256, 0, stream>>>(tbuf, 256, 0, data, 10, 0, NN, 2);
  lin_selu_k<<<cdiv(NN * 254, 256), 256, 0, stream>>>(data, W_lin, b_lin, tbuf,
                                                      NN, 10, 254, 256, 2);

  // layer 1: 256 -> 6x508
  run_gat_layer(tbuf, 256, W1, a1s, a1d, 508, coords0, c1, fbuf, true,
                wpbuf, xbuf, aggbuf, asb, adb, mbuf, denbuf, ebuf, ncoord,
                src, dst, bd, stream);
  // t2 = [c1, coords, f1]  (width 512)
  copy_cols_k<<<cdiv(NN * 2, 256), 256, 0, stream>>>(tbuf, 512, 0, c1, 2, 0, NN, 2);
  copy_cols_k<<<cdiv(NN * 2, 256), 256, 0, stream>>>(tbuf, 512, 2, coords0, 2, 0, NN, 2);
  copy_cols_k<<<cdiv(NN * 508, 256), 256, 0, stream>>>(tbuf, 512, 4, fbuf, 508, 0, NN, 508);

  // layer 2: 512 -> 6x250
  run_gat_layer(tbuf, 512, W2, a2s, a2d, 250, c1, c2, fbuf, true,
                wpbuf, xbuf, aggbuf, asb, adb, mbuf, denbuf, ebuf, ncoord,
                src, dst, bd, stream);
  // t3 = [c2, c1, coords, f2]  (width 256)
  copy_cols_k<<<cdiv(NN * 2, 256), 256, 0, stream>>>(tbuf, 256, 0, c2, 2, 0, NN, 2);
  copy_cols_k<<<cdiv(NN * 2, 256), 256, 0, stream>>>(tbuf, 256, 2, c1, 2, 0, NN, 2);
  copy_cols_k<<<cdiv(NN * 2, 256), 256, 0, stream>>>(tbuf, 256, 4, coords0, 2, 0, NN, 2);
  copy_cols_k<<<cdiv(NN * 250, 256), 256, 0, stream>>>(tbuf, 256, 6, fbuf, 250, 0, NN, 250);

  // layer 3: 256 -> 6x120
  run_gat_layer(tbuf, 256, W3, a3s, a3d, 120, c2, c3, fbuf, true,
                wpbuf, xbuf, aggbuf, asb, adb, mbuf, denbuf, ebuf, ncoord,
                src, dst, bd, stream);
  // t4 = [c3, c2, c1, coords, f3]  (width 128)
  copy_cols_k<<<cdiv(NN * 2, 256), 256, 0, stream>>>(tbuf, 128, 0, c3, 2, 0, NN, 2);
  copy_cols_k<<<cdiv(NN * 2, 256), 256, 0, stream>>>(tbuf, 128, 2, c2, 2, 0, NN, 2);
  copy_cols_k<<<cdiv(NN * 2, 256), 256, 0, stream>>>(tbuf, 128, 4, c1, 2, 0, NN, 2);
  copy_cols_k<<<cdiv(NN * 2, 256), 256, 0, stream>>>(tbuf, 128, 6, coords0, 2, 0, NN, 2);
  copy_cols_k<<<cdiv(NN * 120, 256), 256, 0, stream>>>(tbuf, 128, 8, fbuf, 120, 0, NN, 120);

  // layer 4: 128 -> 6x20 ; only c4 is needed -> write coords straight to d_out
  run_gat_layer(tbuf, 128, W4, a4s, a4d, 20, c3, (float*)d_out, fbuf, false,
                wpbuf, xbuf, aggbuf, asb, adb, mbuf, denbuf, ebuf, ncoord,
                src, dst, bd, stream);
}